// Decoder_24584392802419
// MI455X (gfx1250) — compile-verified
//
#include <hip/hip_runtime.h>

// Problem constants (from reference): B=32, S=100, IN=4, H=128, SS=128
#define BB 32
#define SS_ 100
#define INF_ 4
#define HH 128
#define ZZ 128
#define TSTEPS 99

typedef float v2f __attribute__((ext_vector_type(2)));
typedef float v8f __attribute__((ext_vector_type(8)));

// ---------------------------------------------------------------------------
// f32 WMMA 16x16 tile, generic row-major B (used by parallel precompute).
// Layouts per CDNA5 ISA 7.12.2:
//   A 16x4 f32 : lane L -> row m=L%16 ; VGPR i holds K = 4*kstep + 2*(L/16) + i
//   B 4x16 f32 : lane L -> col n=L%16 ; VGPR i holds K = 4*kstep + 2*(L/16) + i
//   C/D 16x16  : lane L -> col n=L%16 ; VGPR i holds row m = i + 8*(L/16)
// ---------------------------------------------------------------------------
__device__ __forceinline__ void wmma_tile_f32(const float* __restrict__ A, int lda,
                                              const float* __restrict__ Bm, int ldb,
                                              int K, v8f& c, int lane) {
  const int half = lane >> 4;
  const int mn   = lane & 15;
  const float* Ar = A + mn * lda + 2 * half;
  const float* Bc = Bm + mn + 2 * half * ldb;
  for (int k0 = 0; k0 < K; k0 += 4) {
    v2f a, b;
    a.x = Ar[k0];
    a.y = Ar[k0 + 1];
    b.x = Bc[k0 * ldb];
    b.y = Bc[k0 * ldb + ldb];
    c = __builtin_amdgcn_wmma_f32_16x16x4_f32(false, a, false, b, (short)0, c,
                                              false, false);
  }
}

// Packed-B variant: B stored as (K/2, N, 2) so each lane's (B[ka][n],B[ka+1][n])
// pair is one contiguous 8-byte load (b64) instead of two strided b32 loads.
__device__ __forceinline__ void wmma_tile_f32_pk(const float* __restrict__ A, int lda,
                                                 const float* __restrict__ Bp, int N,
                                                 int K, v8f& c, int lane) {
  const int half = lane >> 4;
  const int mn   = lane & 15;
  const float* Ar = A + mn * lda + 2 * half;
  const float* Bq = Bp + half * 2 * N + 2 * mn;
  for (int k0 = 0; k0 < K; k0 += 4) {
    v2f a, b;
    a.x = Ar[k0];
    a.y = Ar[k0 + 1];
    b.x = Bq[k0 * N];      // (kk = k0/2 + half, n) pair, contiguous
    b.y = Bq[k0 * N + 1];
    c = __builtin_amdgcn_wmma_f32_16x16x4_f32(false, a, false, b, (short)0, c,
                                              false, false);
  }
}

__device__ __forceinline__ void wmma_store_f32(float* D, int ldd, const v8f& c,
                                               int lane) {
  const int half = lane >> 4;
  const int n = lane & 15;
#pragma unroll
  for (int i = 0; i < 8; ++i) D[(i + 8 * half) * ldd + n] = c[i];
}

// ---------------------------------------------------------------------------
// P1: ref_h_all[(t*B+b)*H + h] = instance[b, sol[b,t]] @ emb_W + emb_b
// ---------------------------------------------------------------------------
__global__ __launch_bounds__(256) void precompute_refh(
    const float* __restrict__ inst, const int* __restrict__ sol,
    const float* __restrict__ embW, const float* __restrict__ embB,
    float* __restrict__ refh) {
  int idx = blockIdx.x * blockDim.x + threadIdx.x;
  if (idx >= TSTEPS * BB * HH) return;
  int h = idx & (HH - 1);
  int r = idx / HH;
  int b = r & (BB - 1);
  int t = r / BB;
  int node = sol[b * SS_ + t];
  const float* x = inst + (size_t)(b * SS_ + node) * INF_;
  float acc = embB[h];
#pragma unroll
  for (int i = 0; i < INF_; ++i) acc += x[i] * embW[i * HH + h];
  refh[idx] = acc;
}

// ---------------------------------------------------------------------------
// P2: three WMMA GEMMs, one 16x16 output tile per wave (grid-stride):
//   G0: gi  (3168 x 384) = refh @ Wih + bih
//   G1: IHa (3200 x 128) = inst_hidden @ attn_W[0:128,:]
//   G2: IHp (3200 x 128) = inst_hidden @ ptr_W [0:128,:]
// ---------------------------------------------------------------------------
__global__ __launch_bounds__(256) void precompute_gemms(
    const float* __restrict__ refh, const float* __restrict__ Wih,
    const float* __restrict__ bih, const float* __restrict__ ihid,
    const float* __restrict__ attnW, const float* __restrict__ ptrW,
    float* __restrict__ gi, float* __restrict__ iha, float* __restrict__ ihp) {
  const int lane = threadIdx.x & 31;
  const int wid = threadIdx.x >> 5;
  const int waves_per_blk = blockDim.x >> 5;
  const int nwaves = gridDim.x * waves_per_blk;
  const int T0 = 198 * 24, T1 = 200 * 8, T2 = 200 * 8;
  for (int t = blockIdx.x * waves_per_blk + wid; t < T0 + T1 + T2; t += nwaves) {
    v8f c = {};
    if (t < T0) {
      int mt = t / 24, nt = t % 24;
      float bias = bih[nt * 16 + (lane & 15)];
#pragma unroll
      for (int i = 0; i < 8; ++i) c[i] = bias;
      wmma_tile_f32(refh + mt * 16 * HH, HH, Wih + nt * 16, 3 * HH, HH, c, lane);
      wmma_store_f32(gi + mt * 16 * (3 * HH) + nt * 16, 3 * HH, c, lane);
    } else if (t < T0 + T1) {
      int tt = t - T0, mt = tt / 8, nt = tt % 8;
      wmma_tile_f32(ihid + mt * 16 * HH, HH, attnW + nt * 16, HH, HH, c, lane);
      wmma_store_f32(iha + mt * 16 * HH + nt * 16, HH, c, lane);
    } else {
      int tt = t - T0 - T1, mt = tt / 8, nt = tt % 8;
      wmma_tile_f32(ihid + mt * 16 * HH, HH, ptrW + nt * 16, HH, HH, c, lane);
      wmma_store_f32(ihp + mt * 16 * HH + nt * 16, HH, c, lane);
    }
  }
}

// ---------------------------------------------------------------------------
// P3: repack decode-path weights into WMMA fragment order (K/2, N, 2).
// ---------------------------------------------------------------------------
__device__ __forceinline__ void pack_one(const float* __restrict__ src, int N,
                                         float* __restrict__ dst, int idx) {
  int j = idx & 1;
  int rest = idx >> 1;
  int n = rest % N;
  int kk = rest / N;
  dst[idx] = src[(2 * kk + j) * N + n];
}

__global__ __launch_bounds__(256) void pack_weights(
    const float* __restrict__ Whh, const float* __restrict__ fc1W,
    const float* __restrict__ fc2W, float* __restrict__ whhP,
    float* __restrict__ fc1P, float* __restrict__ fc2P) {
  int idx = blockIdx.x * blockDim.x + threadIdx.x;
  const int C0 = HH * 3 * HH;      // 49152
  const int C1 = 3 * HH * 2 * HH;  // 98304
  const int C2 = 2 * HH * HH;      // 32768
  if (idx < C0) pack_one(Whh, 3 * HH, whhP, idx);
  else if (idx < C0 + C1) pack_one(fc1W, 2 * HH, fc1P, idx - C0);
  else if (idx < C0 + C1 + C2) pack_one(fc2W, HH, fc2P, idx - C0 - C1);
}

// ---------------------------------------------------------------------------
// Decode: single persistent workgroup (1024 threads = 32 wave32 on one WGP).
// All per-step activations + the two small weight matrices live in LDS;
// the big weights stream (packed) from L2 each step.
// ---------------------------------------------------------------------------
__global__ __launch_bounds__(1024) void decode_kernel(
    const float* __restrict__ ihid, const int* __restrict__ sol,
    const float* __restrict__ Zg, const float* __restrict__ whhP,
    const float* __restrict__ bhh, const float* __restrict__ attnW,
    const float* __restrict__ attnV, const float* __restrict__ ptrW,
    const float* __restrict__ ptrV, const float* __restrict__ fc1P,
    const float* __restrict__ fc1b, const float* __restrict__ fc2P,
    const float* __restrict__ fc2b, const float* __restrict__ refh,
    const float* __restrict__ gi, const float* __restrict__ IHa,
    const float* __restrict__ IHp, float* __restrict__ out) {
  __shared__ __align__(16) float hh[BB * HH];        // GRU state        16 KB
  __shared__ __align__(16) float buf1[BB * 3 * HH];  // gh/fcin/gptr     48 KB
  __shared__ __align__(16) float buf2[BB * 2 * HH];  // dec/g1           32 KB
  __shared__ __align__(16) float ctx[BB * HH];       // context/g        16 KB
  __shared__ __align__(16) float ea[BB * SS_];       // e/a/masked       12.5 KB
  __shared__ __align__(16) float sZ[BB * ZZ];        // Z                16 KB
  __shared__ __align__(16) float sA2[HH * HH];       // attn_W[128:] pkd 64 KB
  __shared__ __align__(16) float sP2[HH * HH];       // ptr_W [128:] pkd 64 KB
  __shared__ __align__(16) float sAv[HH], sPv[HH];

  const int tid = threadIdx.x;
  const int lane = tid & 31;
  const int wid = tid >> 5;  // 0..31, one wave per batch row

  for (int i = tid; i < BB * HH; i += 1024) hh[i] = 0.f;
  for (int i = tid; i < BB * ZZ; i += 1024) sZ[i] = Zg[i];
  // Pack attn_W[128:,:] and ptr_W[128:,:] fragments into LDS (once).
  for (int i = tid; i < HH * HH; i += 1024) {
    int j = i & 1;
    int rest = i >> 1;
    int n = rest & (HH - 1);
    int kk = rest >> 7;
    sA2[i] = attnW[(HH + 2 * kk + j) * HH + n];
    sP2[i] = ptrW[(HH + 2 * kk + j) * HH + n];
  }
  if (tid < HH) { sAv[tid] = attnV[tid]; sPv[tid] = ptrV[tid]; }
  if (tid < BB) out[tid * SS_] = (float)sol[tid * SS_];  // tour_idx[:,0]
  __syncthreads();

  for (int t = 0; t < TSTEPS; ++t) {
    const float* gi_t = gi + (size_t)(t * BB) * (3 * HH);

    // -- 1: gh = hh @ Whh + bhh  -> buf1 (32x384), 48 WMMA tiles
    for (int tile = wid; tile < 48; tile += 32) {
      int mt = tile / 24, nt = tile % 24;
      v8f c;
      float bias = bhh[nt * 16 + (lane & 15)];
#pragma unroll
      for (int i = 0; i < 8; ++i) c[i] = bias;
      wmma_tile_f32_pk(hh + mt * 16 * HH, HH, whhP + 2 * nt * 16, 3 * HH, HH, c, lane);
      wmma_store_f32(buf1 + mt * 16 * (3 * HH) + nt * 16, 3 * HH, c, lane);
    }
    __syncthreads();

    // -- 2: GRU gates (elementwise), hh in place (thread-owned slots)
    for (int p = tid; p < BB * HH; p += 1024) {
      int b = p >> 7, h = p & (HH - 1);
      float gir = gi_t[b * 3 * HH + h];
      float giz = gi_t[b * 3 * HH + HH + h];
      float gin = gi_t[b * 3 * HH + 2 * HH + h];
      float ghr = buf1[b * 3 * HH + h];
      float ghz = buf1[b * 3 * HH + HH + h];
      float ghn = buf1[b * 3 * HH + 2 * HH + h];
      float r = 1.f / (1.f + expf(-(gir + ghr)));
      float z = 1.f / (1.f + expf(-(giz + ghz)));
      float n = tanhf(gin + r * ghn);
      hh[p] = (1.f - z) * n + z * hh[p];
    }
    __syncthreads();

    // -- 3: dec = hh @ attn_W[128:,:] (LDS-resident packed) -> buf2[0:32*128]
    for (int tile = wid; tile < 16; tile += 32) {
      int mt = tile / 8, nt = tile % 8;
      v8f c = {};
      wmma_tile_f32_pk(hh + mt * 16 * HH, HH, sA2 + 2 * nt * 16, HH, HH, c, lane);
      wmma_store_f32(buf2 + mt * 16 * HH + nt * 16, HH, c, lane);
    }
    __syncthreads();

    // -- 4: e[b,s] = sum_h tanh(IHa + dec) * attn_v   (float4 vectorized)
    for (int p = tid; p < BB * SS_; p += 1024) {
      int b = p / SS_, s = p - b * SS_;
      const float4* ih4 = (const float4*)(IHa + (size_t)(b * SS_ + s) * HH);
      const float4* dc4 = (const float4*)(buf2 + b * HH);
      const float4* av4 = (const float4*)sAv;
      float acc = 0.f;
      for (int q = 0; q < HH / 4; ++q) {
        float4 iv = ih4[q], dv = dc4[q], vv = av4[q];
        acc += tanhf(iv.x + dv.x) * vv.x;
        acc += tanhf(iv.y + dv.y) * vv.y;
        acc += tanhf(iv.z + dv.z) * vv.z;
        acc += tanhf(iv.w + dv.w) * vv.w;
      }
      ea[p] = acc;
    }
    __syncthreads();

    // -- 5: a = softmax_s(e) (unmasked), wave per batch row
    {
      int b = wid;
      float v0[4];
      float mx = -1e30f;
#pragma unroll
      for (int i = 0; i < 4; ++i) {
        int s = lane + 32 * i;
        v0[i] = (s < SS_) ? ea[b * SS_ + s] : -1e30f;
        mx = fmaxf(mx, v0[i]);
      }
      for (int off = 16; off; off >>= 1) mx = fmaxf(mx, __shfl_xor(mx, off, 32));
      float sum = 0.f;
#pragma unroll
      for (int i = 0; i < 4; ++i) {
        v0[i] = (lane + 32 * i < SS_) ? expf(v0[i] - mx) : 0.f;
        sum += v0[i];
      }
      for (int off = 16; off; off >>= 1) sum += __shfl_xor(sum, off, 32);
      float inv = 1.f / sum;
#pragma unroll
      for (int i = 0; i < 4; ++i) {
        int s = lane + 32 * i;
        if (s < SS_) ea[b * SS_ + s] = v0[i] * inv;
      }
    }
    __syncthreads();

    // -- 6: context[b,h0..h0+3] = sum_s a * instance_hidden  (float4, 1 item/thread)
    {
      int b = tid >> 5;
      int h0 = (tid & 31) * 4;
      const float4* ih4 = (const float4*)(ihid + (size_t)b * SS_ * HH + h0);
      const float* ab = ea + b * SS_;
      float ax = 0.f, ay = 0.f, az = 0.f, aw = 0.f;
      for (int s = 0; s < SS_; ++s) {
        float f = ab[s];
        float4 v = ih4[s * (HH / 4)];
        ax += f * v.x; ay += f * v.y; az += f * v.z; aw += f * v.w;
      }
      float4* cd = (float4*)(ctx + b * HH + h0);
      float4 o; o.x = ax; o.y = ay; o.z = az; o.w = aw;
      *cd = o;
    }
    __syncthreads();

    // -- 7: fcin = [context | Z | ref_h(t)] -> buf1
    {
      const float* rh = refh + (size_t)(t * BB) * HH;
      for (int p = tid; p < BB * HH; p += 1024) {
        int b = p >> 7, h = p & (HH - 1);
        buf1[b * 3 * HH + h] = ctx[p];
        buf1[b * 3 * HH + HH + h] = sZ[p];
        buf1[b * 3 * HH + 2 * HH + h] = rh[b * HH + h];
      }
    }
    __syncthreads();

    // -- 8: g1 = fcin @ fc1_W + fc1_b -> buf2 (32x256), 32 tiles, K=384
    for (int tile = wid; tile < 32; tile += 32) {
      int mt = tile / 16, nt = tile % 16;
      v8f c;
      float bias = fc1b[nt * 16 + (lane & 15)];
#pragma unroll
      for (int i = 0; i < 8; ++i) c[i] = bias;
      wmma_tile_f32_pk(buf1 + mt * 16 * 3 * HH, 3 * HH, fc1P + 2 * nt * 16, 2 * HH,
                       3 * HH, c, lane);
      wmma_store_f32(buf2 + mt * 16 * 2 * HH + nt * 16, 2 * HH, c, lane);
    }
    __syncthreads();

    // -- 9: g = g1 @ fc2_W + fc2_b -> ctx (32x128), 16 tiles, K=256
    for (int tile = wid; tile < 16; tile += 32) {
      int mt = tile / 8, nt = tile % 8;
      v8f c;
      float bias = fc2b[nt * 16 + (lane & 15)];
#pragma unroll
      for (int i = 0; i < 8; ++i) c[i] = bias;
      wmma_tile_f32_pk(buf2 + mt * 16 * 2 * HH, 2 * HH, fc2P + 2 * nt * 16, HH,
                       2 * HH, c, lane);
      wmma_store_f32(ctx + mt * 16 * HH + nt * 16, HH, c, lane);
    }
    __syncthreads();

    // -- 10: gptr = g @ ptr_W[128:,:] (LDS-resident packed) -> buf1[0:32*128]
    for (int tile = wid; tile < 16; tile += 32) {
      int mt = tile / 8, nt = tile % 8;
      v8f c = {};
      wmma_tile_f32_pk(ctx + mt * 16 * HH, HH, sP2 + 2 * nt * 16, HH, HH, c, lane);
      wmma_store_f32(buf1 + mt * 16 * HH + nt * 16, HH, c, lane);
    }
    __syncthreads();

    // -- 11: masked logits -> ea  (log_mask = -1e9 at solution[:,0], constant)
    for (int p = tid; p < BB * SS_; p += 1024) {
      int b = p / SS_, s = p - b * SS_;
      const float4* ih4 = (const float4*)(IHp + (size_t)(b * SS_ + s) * HH);
      const float4* gp4 = (const float4*)(buf1 + b * HH);
      const float4* pv4 = (const float4*)sPv;
      float acc = 0.f;
      for (int q = 0; q < HH / 4; ++q) {
        float4 iv = ih4[q], gv = gp4[q], vv = pv4[q];
        acc += tanhf(iv.x + gv.x) * vv.x;
        acc += tanhf(iv.y + gv.y) * vv.y;
        acc += tanhf(iv.z + gv.z) * vv.z;
        acc += tanhf(iv.w + gv.w) * vv.w;
      }
      if (s == sol[b * SS_]) acc += -1e9f;
      ea[p] = acc;
    }
    __syncthreads();

    // -- 12: argmax (first-index tie) + log_softmax at target, wave per b
    {
      int b = wid;
      int ptr_t = sol[b * SS_ + t + 1];
      float v0[4];
      float mx = -1e30f;
      int ax = 0;
#pragma unroll
      for (int i = 0; i < 4; ++i) {
        int s = lane + 32 * i;
        v0[i] = (s < SS_) ? ea[b * SS_ + s] : -3e30f;
        if (v0[i] > mx) { mx = v0[i]; ax = s; }
      }
      for (int off = 16; off; off >>= 1) {
        float omx = __shfl_xor(mx, off, 32);
        int oax = __shfl_xor(ax, off, 32);
        if (omx > mx || (omx == mx && oax < ax)) { mx = omx; ax = oax; }
      }
      float sum = 0.f, vt = 0.f;
#pragma unroll
      for (int i = 0; i < 4; ++i) {
        int s = lane + 32 * i;
        if (s < SS_) {
          sum += expf(v0[i] - mx);
          if (s == ptr_t) vt = v0[i];
        }
      }
      for (int off = 16; off; off >>= 1) {
        sum += __shfl_xor(sum, off, 32);
        vt += __shfl_xor(vt, off, 32);
      }
      if (lane == 0) {
        out[b * SS_ + t + 1] = (float)ax;                         // tour_idx
        out[BB * SS_ + b * (SS_ - 1) + t] = vt - mx - logf(sum);  // tour_logp
      }
    }
    __syncthreads();
  }
}

// ---------------------------------------------------------------------------
extern "C" void kernel_launch(void* const* d_in, const int* in_sizes, int n_in,
                              void* d_out, int out_size, void* d_ws,
                              size_t ws_size, hipStream_t stream) {
  const float* inst = (const float*)d_in[0];
  const int* sol    = (const int*)d_in[1];
  const float* Z    = (const float*)d_in[2];
  const float* ihid = (const float*)d_in[3];
  const float* embW = (const float*)d_in[4];
  const float* embB = (const float*)d_in[5];
  const float* Wih  = (const float*)d_in[6];
  const float* Whh  = (const float*)d_in[7];
  const float* bih  = (const float*)d_in[8];
  const float* bhh  = (const float*)d_in[9];
  const float* attnW = (const float*)d_in[10];
  const float* attnV = (const float*)d_in[11];
  const float* ptrW  = (const float*)d_in[12];
  const float* ptrV  = (const float*)d_in[13];
  const float* fc1W  = (const float*)d_in[14];
  const float* fc1b  = (const float*)d_in[15];
  const float* fc2W  = (const float*)d_in[16];
  const float* fc2b  = (const float*)d_in[17];

  float* ws = (float*)d_ws;
  float* refh = ws;                                   // 405504
  float* gi   = refh + (size_t)TSTEPS * BB * HH;      // 1216512
  float* iha  = gi + (size_t)TSTEPS * BB * 3 * HH;    // 409600
  float* ihp  = iha + (size_t)BB * SS_ * HH;          // 409600
  float* whhP = ihp + (size_t)BB * SS_ * HH;          // 49152
  float* fc1P = whhP + (size_t)HH * 3 * HH;           // 98304
  float* fc2P = fc1P + (size_t)3 * HH * 2 * HH;       // 32768
  float* out  = (float*)d_out;

  int n1 = TSTEPS * BB * HH;
  precompute_refh<<<(n1 + 255) / 256, 256, 0, stream>>>(inst, sol, embW, embB, refh);
  precompute_gemms<<<994, 256, 0, stream>>>(refh, Wih, bih, ihid, attnW, ptrW,
                                            gi, iha, ihp);
  int npack = HH * 3 * HH + 3 * HH * 2 * HH + 2 * HH * HH;
  pack_weights<<<(npack + 255) / 256, 256, 0, stream>>>(Whh, fc1W, fc2W,
                                                        whhP, fc1P, fc2P);
  decode_kernel<<<1, 1024, 0, stream>>>(ihid, sol, Z, whhP, bhh, attnW, attnV,
                                        ptrW, ptrV, fc1P, fc1b, fc2P, fc2b,
                                        refh, gi, iha, ihp, out);
}